// DepthNet_88579405513057
// MI455X (gfx1250) — compile-verified
//
#include <hip/hip_runtime.h>
#include <math.h>

typedef float v2f __attribute__((ext_vector_type(2)));
typedef float v8f __attribute__((ext_vector_type(8)));

#define HS 256
#define WSZ 256
#define HT 128
#define WT 128
#define CCH 32
#define NVIEW 3
#define ND 48
#define NPIX (HT * WT)
#define HSWS (HS * WSZ)

// ---------------------------------------------------------------------------
// Kernel A: proj[v] = (src_ints[v] @ src_exts[v][:3,:]) @ inv(tar_proj)
// tar_proj = [[tar_ints @ tar_exts[:3,:]]; [0,0,0,1]]
// One thread; tiny. Writes 3 x (3x4) = 36 floats into ws.
// ---------------------------------------------------------------------------
__global__ void dn_setup_proj(const float* __restrict__ src_exts,
                              const float* __restrict__ src_ints,
                              const float* __restrict__ tar_exts,
                              const float* __restrict__ tar_ints,
                              float* __restrict__ ws_proj) {
  if (threadIdx.x != 0 || blockIdx.x != 0) return;

  float m[16];
  for (int r = 0; r < 3; ++r)
    for (int c = 0; c < 4; ++c) {
      float s = 0.f;
      for (int k = 0; k < 3; ++k) s += tar_ints[r * 3 + k] * tar_exts[k * 4 + c];
      m[r * 4 + c] = s;
    }
  m[12] = 0.f; m[13] = 0.f; m[14] = 0.f; m[15] = 1.f;

  // 4x4 inverse via cofactors (gluInvertMatrix)
  float inv[16];
  inv[0]  =  m[5]*m[10]*m[15] - m[5]*m[11]*m[14] - m[9]*m[6]*m[15] + m[9]*m[7]*m[14] + m[13]*m[6]*m[11] - m[13]*m[7]*m[10];
  inv[4]  = -m[4]*m[10]*m[15] + m[4]*m[11]*m[14] + m[8]*m[6]*m[15] - m[8]*m[7]*m[14] - m[12]*m[6]*m[11] + m[12]*m[7]*m[10];
  inv[8]  =  m[4]*m[9]*m[15]  - m[4]*m[11]*m[13] - m[8]*m[5]*m[15] + m[8]*m[7]*m[13] + m[12]*m[5]*m[11] - m[12]*m[7]*m[9];
  inv[12] = -m[4]*m[9]*m[14]  + m[4]*m[10]*m[13] + m[8]*m[5]*m[14] - m[8]*m[6]*m[13] - m[12]*m[5]*m[10] + m[12]*m[6]*m[9];
  inv[1]  = -m[1]*m[10]*m[15] + m[1]*m[11]*m[14] + m[9]*m[2]*m[15] - m[9]*m[3]*m[14] - m[13]*m[2]*m[11] + m[13]*m[3]*m[10];
  inv[5]  =  m[0]*m[10]*m[15] - m[0]*m[11]*m[14] - m[8]*m[2]*m[15] + m[8]*m[3]*m[14] + m[12]*m[2]*m[11] - m[12]*m[3]*m[10];
  inv[9]  = -m[0]*m[9]*m[15]  + m[0]*m[11]*m[13] + m[8]*m[1]*m[15] - m[8]*m[3]*m[13] - m[12]*m[1]*m[11] + m[12]*m[3]*m[9];
  inv[13] =  m[0]*m[9]*m[14]  - m[0]*m[10]*m[13] - m[8]*m[1]*m[14] + m[8]*m[2]*m[13] + m[12]*m[1]*m[10] - m[12]*m[2]*m[9];
  inv[2]  =  m[1]*m[6]*m[15]  - m[1]*m[7]*m[14]  - m[5]*m[2]*m[15] + m[5]*m[3]*m[14] + m[13]*m[2]*m[7]  - m[13]*m[3]*m[6];
  inv[6]  = -m[0]*m[6]*m[15]  + m[0]*m[7]*m[14]  + m[4]*m[2]*m[15] - m[4]*m[3]*m[14] - m[12]*m[2]*m[7]  + m[12]*m[3]*m[6];
  inv[10] =  m[0]*m[5]*m[15]  - m[0]*m[7]*m[13]  - m[4]*m[1]*m[15] + m[4]*m[3]*m[13] + m[12]*m[1]*m[7]  - m[12]*m[3]*m[5];
  inv[14] = -m[0]*m[5]*m[14]  + m[0]*m[6]*m[13]  + m[4]*m[1]*m[14] - m[4]*m[2]*m[13] - m[12]*m[1]*m[6]  + m[12]*m[2]*m[5];
  inv[3]  = -m[1]*m[6]*m[11]  + m[1]*m[7]*m[10]  + m[5]*m[2]*m[11] - m[5]*m[3]*m[10] - m[9]*m[2]*m[7]   + m[9]*m[3]*m[6];
  inv[7]  =  m[0]*m[6]*m[11]  - m[0]*m[7]*m[10]  - m[4]*m[2]*m[11] + m[4]*m[3]*m[10] + m[8]*m[2]*m[7]   - m[8]*m[3]*m[6];
  inv[11] = -m[0]*m[5]*m[11]  + m[0]*m[7]*m[9]   + m[4]*m[1]*m[11] - m[4]*m[3]*m[9]  - m[8]*m[1]*m[7]   + m[8]*m[3]*m[5];
  inv[15] =  m[0]*m[5]*m[10]  - m[0]*m[6]*m[9]   - m[4]*m[1]*m[10] + m[4]*m[2]*m[9]  + m[8]*m[1]*m[6]   - m[8]*m[2]*m[5];
  float det = m[0]*inv[0] + m[1]*inv[4] + m[2]*inv[8] + m[3]*inv[12];
  float idet = 1.0f / det;
  for (int i = 0; i < 16; ++i) inv[i] *= idet;

  for (int v = 0; v < NVIEW; ++v) {
    float SP[12];
    for (int r = 0; r < 3; ++r)
      for (int c = 0; c < 4; ++c) {
        float s = 0.f;
        for (int k = 0; k < 3; ++k)
          s += src_ints[v * 9 + r * 3 + k] * src_exts[v * 16 + k * 4 + c];
        SP[r * 4 + c] = s;
      }
    for (int r = 0; r < 3; ++r)
      for (int c = 0; c < 4; ++c) {
        float s = 0.f;
        for (int k = 0; k < 4; ++k) s += SP[r * 4 + k] * inv[k * 4 + c];
        ws_proj[v * 12 + r * 4 + c] = s;
      }
  }
}

// ---------------------------------------------------------------------------
// Kernel B: cost volume. One thread per (depth, target pixel).
// cost(d,p) = S2/(V*C) - S11/(V^2*C),  S2 = sum_{v,c} w^2, S11 = sum_c (sum_v w)^2
// Tap validity folded into bilinear weights; addresses clamped -> no branches.
// ---------------------------------------------------------------------------
__global__ void dn_cost_volume(const float* __restrict__ feat,
                               const float* __restrict__ nearfar,
                               const float* __restrict__ proj,
                               float* __restrict__ cost) {
  int gid = blockIdx.x * blockDim.x + threadIdx.x;
  if (gid >= ND * NPIX) return;
  int d = gid / NPIX;
  int p = gid - d * NPIX;
  int py = p >> 7;
  int px = p & (WT - 1);
  float gx = (float)px + 0.5f;
  float gy = (float)py + 0.5f;

  float nearv = nearfar[p];
  float farv  = nearfar[NPIX + p];
  float dv = nearv + (farv - nearv) * ((float)d * (1.0f / (float)(ND - 1)));

  int   off[NVIEW][4];
  float wgt[NVIEW][4];
#pragma unroll
  for (int v = 0; v < NVIEW; ++v) {
    const float* P = proj + v * 12;
    float bx = P[0] * gx + P[1] * gy + P[2];
    float by = P[4] * gx + P[5] * gy + P[6];
    float bz = P[8] * gx + P[9] * gy + P[10];
    float pxs = bx * dv + P[3];
    float pys = by * dv + P[7];
    float pzs = bz * dv + P[11];
    float z  = fmaxf(pzs, 1e-6f);
    float iz = 1.0f / z;
    float sx = fminf(fmaxf(pxs * iz, -1e6f), 1e6f) - 0.5f;
    float sy = fminf(fmaxf(pys * iz, -1e6f), 1e6f) - 0.5f;
    float fx0 = floorf(sx), fy0 = floorf(sy);
    float wx = sx - fx0, wy = sy - fy0;
    int x0 = (int)fx0, y0 = (int)fy0;
    int x1 = x0 + 1, y1 = y0 + 1;
    float vx0 = (x0 >= 0 && x0 < WSZ) ? 1.f : 0.f;
    float vx1 = (x1 >= 0 && x1 < WSZ) ? 1.f : 0.f;
    float vy0 = (y0 >= 0 && y0 < HS) ? 1.f : 0.f;
    float vy1 = (y1 >= 0 && y1 < HS) ? 1.f : 0.f;
    int xc0 = min(max(x0, 0), WSZ - 1);
    int xc1 = min(max(x1, 0), WSZ - 1);
    int yc0 = min(max(y0, 0), HS - 1);
    int yc1 = min(max(y1, 0), HS - 1);
    off[v][0] = yc0 * WSZ + xc0;
    off[v][1] = yc0 * WSZ + xc1;
    off[v][2] = yc1 * WSZ + xc0;
    off[v][3] = yc1 * WSZ + xc1;
    wgt[v][0] = (1.f - wy) * (1.f - wx) * vy0 * vx0;
    wgt[v][1] = (1.f - wy) * wx * vy0 * vx1;
    wgt[v][2] = wy * (1.f - wx) * vy1 * vx0;
    wgt[v][3] = wy * wx * vy1 * vx1;
  }

  const float* f0 = feat;
  const float* f1 = feat + (size_t)1 * CCH * HSWS;
  const float* f2 = feat + (size_t)2 * CCH * HSWS;

  float s2 = 0.f, s11 = 0.f;
#pragma unroll 4
  for (int c = 0; c < CCH; ++c) {
    int cb = c * HSWS;
    float w0 = wgt[0][0] * f0[cb + off[0][0]] + wgt[0][1] * f0[cb + off[0][1]] +
               wgt[0][2] * f0[cb + off[0][2]] + wgt[0][3] * f0[cb + off[0][3]];
    float w1 = wgt[1][0] * f1[cb + off[1][0]] + wgt[1][1] * f1[cb + off[1][1]] +
               wgt[1][2] * f1[cb + off[1][2]] + wgt[1][3] * f1[cb + off[1][3]];
    float w2 = wgt[2][0] * f2[cb + off[2][0]] + wgt[2][1] * f2[cb + off[2][1]] +
               wgt[2][2] * f2[cb + off[2][2]] + wgt[2][3] * f2[cb + off[2][3]];
    float t1 = w0 + w1 + w2;
    s2 += w0 * w0 + w1 * w1 + w2 * w2;
    s11 += t1 * t1;
  }
  cost[gid] = s2 * (1.0f / 96.0f) - s11 * (1.0f / 288.0f);
}

// ---------------------------------------------------------------------------
// Kernel C: softmax over depth + depth/CI regression.
// Per wave: 16 pixels. Moments [Z, sum(e*t), sum(e*t^2)] computed as
// (16 x 48) @ (48 x 3) via 12 chained V_WMMA_F32_16X16X4_F32 (exact f32).
// A 16x4 layout (ISA): lanes 0-15 hold K0/K1, lanes 16-31 hold K2/K3.
// D 16x16 layout: VGPR r = row r (lanes 0-15) / row r+8 (lanes 16-31).
// ---------------------------------------------------------------------------
__global__ void dn_depth_regress(const float* __restrict__ cost,
                                 const float* __restrict__ nearfar,
                                 float* __restrict__ out) {
  __shared__ float moments[128][4];
  int lane = threadIdx.x & 31;
  int wave = threadIdx.x >> 5;
  int pm = lane & 15;     // row (pixel-within-tile) for A, column for B/D
  int half = lane >> 4;   // K half for A/B
  int pixBase = blockIdx.x * 128 + wave * 16;
  int p = pixBase + pm;

  // per-pixel min cost (== max of -cost) for stable softmax
  float mc = 3.4e38f;
  for (int dd = 0; dd < ND; ++dd) mc = fminf(mc, cost[dd * NPIX + p]);

  v8f acc = {0.f, 0.f, 0.f, 0.f, 0.f, 0.f, 0.f, 0.f};
#pragma unroll
  for (int j = 0; j < 12; ++j) {
    int k0 = 4 * j + 2 * half;
    int k1 = k0 + 1;
    v2f a, b;
    a.x = __expf(mc - cost[k0 * NPIX + p]);
    a.y = __expf(mc - cost[k1 * NPIX + p]);
    float t0 = (float)k0 * (1.0f / (float)(ND - 1));
    float t1 = (float)k1 * (1.0f / (float)(ND - 1));
    // column pm: 0 -> 1 (partition Z), 1 -> t, 2 -> t^2, else 0
    b.x = (pm == 0) ? 1.0f : ((pm == 1) ? t0 : ((pm == 2) ? t0 * t0 : 0.0f));
    b.y = (pm == 0) ? 1.0f : ((pm == 1) ? t1 : ((pm == 2) ? t1 * t1 : 0.0f));
    acc = __builtin_amdgcn_wmma_f32_16x16x4_f32(false, a, false, b,
                                                (short)0, acc, false, false);
  }

  // Scatter D tile (pixel, moment) to LDS: lanes with column < 3 are useful.
  if (pm < 3) {
#pragma unroll
    for (int r = 0; r < 8; ++r)
      moments[wave * 16 + half * 8 + r][pm] = acc[r];
  }
  __syncthreads();

  if (lane < 16) {
    int pp = pixBase + lane;
    float Z   = moments[wave * 16 + lane][0];
    float Et  = moments[wave * 16 + lane][1];
    float Et2 = moments[wave * 16 + lane][2];
    float nearv = nearfar[pp];
    float farv  = nearfar[NPIX + pp];
    float iZ = 1.0f / Z;
    float et = Et * iZ;
    float et2 = Et2 * iZ;
    float fn = farv - nearv;
    float depth = nearv + fn * et;
    float var = fn * fn * (et2 - et * et);
    float hci = sqrtf(fmaxf(var, 1e-12f));
    out[pp] = depth;
    out[NPIX + pp] = fmaxf(depth - hci, nearv);      // ci low (clamp to dv[0]=near)
    out[2 * NPIX + pp] = fminf(depth + hci, farv);   // ci high (clamp to dv[D-1]=far)
  }
}

// ---------------------------------------------------------------------------
extern "C" void kernel_launch(void* const* d_in, const int* in_sizes, int n_in,
                              void* d_out, int out_size, void* d_ws, size_t ws_size,
                              hipStream_t stream) {
  (void)in_sizes; (void)n_in; (void)out_size; (void)ws_size;
  const float* src_feat = (const float*)d_in[0];
  const float* src_exts = (const float*)d_in[1];
  const float* src_ints = (const float*)d_in[2];
  const float* tar_exts = (const float*)d_in[3];
  const float* tar_ints = (const float*)d_in[4];
  const float* near_far = (const float*)d_in[5];
  // d_in[6] = num_depth (device scalar); grid shape must be host-side -> ND=48.

  float* ws_proj = (float*)d_ws;            // 36 floats
  float* ws_cost = (float*)d_ws + 64;       // 48*16384 floats (3 MB)
  float* out = (float*)d_out;               // depth(16384) + ci(2*16384)

  dn_setup_proj<<<1, 32, 0, stream>>>(src_exts, src_ints, tar_exts, tar_ints, ws_proj);

  int total = ND * NPIX;
  dn_cost_volume<<<(total + 255) / 256, 256, 0, stream>>>(src_feat, near_far,
                                                          ws_proj, ws_cost);

  dn_depth_regress<<<NPIX / 128, 256, 0, stream>>>(ws_cost, near_far, out);
}